// BayesianVectorRenderer_30597347016923
// MI455X (gfx1250) — compile-verified
//
#include <hip/hip_runtime.h>
#include <hip/hip_bf16.h>

#define H 1024
#define W 1024
#define NPTS 512            // 16 segments * 32 samples
#define NEDGE 512
#define TILE_X 32
#define TILE_Y 8
#define EDGE_FLOATS (5 * NEDGE)   // x0 | y0 | dx | inv_dy | coeff (SoA)
#define CUT 20.0f                 // sigmoid(20) = 1 - 2e-9: saturation cutoff

typedef __attribute__((ext_vector_type(2))) float v2f;
typedef __attribute__((ext_vector_type(8))) float v8f;

// sigmoid(z) = 1 / (1 + exp(-z)) using hardware v_exp_f32 (2^x) + v_rcp_f32
__device__ __forceinline__ float fast_sigmoid(float z) {
    const float LOG2E = 1.4426950408889634f;
    return __builtin_amdgcn_rcpf(1.0f + __builtin_amdgcn_exp2f(-LOG2E * z));
}

__device__ __forceinline__ float bern_w(int j, int k) {
    float t  = (float)j * (1.0f / 31.0f);   // linspace(0,1,32)
    float mt = 1.0f - t;
    switch (k) {
        case 0:  return mt * mt * mt;
        case 1:  return 3.0f * mt * mt * t;
        case 2:  return 3.0f * mt * t * t;
        default: return t * t * t;
    }
}

// ---------------------------------------------------------------------------
// Kernel 1: Bézier sampling as a real GEMM on the WMMA unit.
//   pts(M=32 samples j, N=32 (seg,coord)) = Bern(32x4) @ CP(4x32), K=4.
//   One wave (32 threads), four V_WMMA_F32_16X16X4_F32 tiles.
// ---------------------------------------------------------------------------
__global__ void bezier_sample_wmma(const float* __restrict__ cp,
                                   float* __restrict__ pts) {
    const int l     = threadIdx.x;        // 0..31, one full wave (EXEC all 1s)
    const int mrow  = l & 15;
    const int khalf = (l >> 4) * 2;       // 0 or 2
#if __has_builtin(__builtin_amdgcn_wmma_f32_16x16x4_f32)
    for (int tm = 0; tm < 2; ++tm) {
        const int j = tm * 16 + mrow;
        v2f a; a.x = bern_w(j, khalf); a.y = bern_w(j, khalf + 1);
        for (int tn = 0; tn < 2; ++tn) {
            const int n     = tn * 16 + mrow;     // column this lane owns
            const int seg   = n >> 1;
            const int coord = n & 1;
            v2f b;
            b.x = cp[(3 * seg + khalf)     * 2 + coord];
            b.y = cp[(3 * seg + khalf + 1) * 2 + coord];
            v8f c = {};
            c = __builtin_amdgcn_wmma_f32_16x16x4_f32(
                    false, a, false, b, (short)0, c, false, false);
            #pragma unroll
            for (int r = 0; r < 8; ++r) {
                const int M  = r + ((l >> 4) ? 8 : 0);
                const int jj = tm * 16 + M;             // sample within segment
                pts[(seg * 32 + jj) * 2 + coord] = c[r];
            }
        }
    }
#else
    for (int p = l; p < NPTS; p += 32) {
        const int seg = p >> 5, j = p & 31;
        float x = 0.f, y = 0.f;
        #pragma unroll
        for (int k = 0; k < 4; ++k) {
            const float w = bern_w(j, k);
            x += w * cp[(3 * seg + k) * 2 + 0];
            y += w * cp[(3 * seg + k) * 2 + 1];
        }
        pts[p * 2 + 0] = x;
        pts[p * 2 + 1] = y;
    }
#endif
}

// ---------------------------------------------------------------------------
// Kernel 2: edge table (SoA): x0 | y0 | dx | 1/(dy+1e-8) | coeff
// ---------------------------------------------------------------------------
__global__ void build_edges(const float* __restrict__ pts,
                            float* __restrict__ edge) {
    const int e = blockIdx.x * blockDim.x + threadIdx.x;
    if (e >= NEDGE) return;
    const int   en = (e + 1) & (NEDGE - 1);
    const float x0 = pts[e  * 2 + 0], y0 = pts[e  * 2 + 1];
    const float x1 = pts[en * 2 + 0], y1 = pts[en * 2 + 1];
    const float dx = x1 - x0;
    const float dy = y1 - y0;
    const float sgn   = (dy > 0.f ? 1.f : 0.f) - (dy < 0.f ? 1.f : 0.f);
    const float coeff = (fabsf(dy) >= 1e-6f) ? sgn : 0.f;
    edge[0 * NEDGE + e] = x0;
    edge[1 * NEDGE + e] = y0;
    edge[2 * NEDGE + e] = dx;
    edge[3 * NEDGE + e] = 1.0f / (dy + 1e-8f);
    edge[4 * NEDGE + e] = coeff;
}

// ---------------------------------------------------------------------------
// Kernel 3: render. Tile = 32 cols x 8 rows, 256 threads = 8 wave32s,
// one wave per image row of the tile.
// Phase 0: async-stage 10KB edge table into LDS (CDNA5 async-to-LDS path).
// Phase 1: wave r scans all 512 edges for its row: computes w = coeff*valid_t
//          and x_cross, then band-classifies against the tile's x-range:
//            left  (xc <= xlo): sigmoid ~ 0   -> dropped
//            right (xc >= xhi): sigmoid ~ 1   -> folded into scalar base[r]
//                               (deterministic shfl_xor tree reduction)
//            active           : compacted in edge order via ballot+popcount
//          Deterministic: no FP atomics, fixed reduction tree, fixed order.
// Phase 2: acc = base[r] + sum over compacted band edges (~7% of 512),
//          one broadcast ds_load_b64 of packed (xc,w) + exp + rcp per edge.
// ---------------------------------------------------------------------------
#if __has_builtin(__builtin_amdgcn_global_load_async_to_lds_b32) && \
    __has_builtin(__builtin_amdgcn_s_wait_asynccnt)
#define USE_ASYNC_LDS 1
#else
#define USE_ASYNC_LDS 0
#endif

__global__ __launch_bounds__(256) void render_winding(
        const float* __restrict__ edge,
        const float* __restrict__ color,
        float4* __restrict__ out) {
    __shared__ float  sE[EDGE_FLOATS];          // 10 KB staged edge table
    __shared__ float2 sAct[TILE_Y][NEDGE];      // 32 KB compacted (xc, w)
    __shared__ float  sBase[TILE_Y];
    __shared__ int    sCnt[TILE_Y];

    const int tid   = threadIdx.x;
    const int lane  = tid & 31;
    const int wv    = tid >> 5;                 // wave id == row in tile
    const int y0blk = blockIdx.y * TILE_Y;
    const int x0blk = blockIdx.x * TILE_X;

#if USE_ASYNC_LDS
    {
        typedef __attribute__((address_space(1))) int* gip;
        typedef __attribute__((address_space(3))) int* lip;
        for (int i = tid; i < EDGE_FLOATS; i += 256)
            __builtin_amdgcn_global_load_async_to_lds_b32(
                (gip)(edge + i), (lip)(sE + i), 0, 0);
        __builtin_amdgcn_s_wait_asynccnt(0);
    }
#else
    for (int i = tid; i < EDGE_FLOATS; i += 256) sE[i] = edge[i];
#endif
    __syncthreads();

    // ---- Phase 1: per-row edge classification + ordered compaction ----
    {
        const int   r   = wv;
        const float yy  = (float)(y0blk + r);
        const float xlo = (float)x0blk - CUT;
        const float xhi = (float)(x0blk + TILE_X - 1) + CUT;
        const float LOG2E = 1.4426950408889634f;
        int   cnt  = 0;
        float base = 0.0f;
        for (int ch = 0; ch < NEDGE / 32; ++ch) {
            const int   e   = ch * 32 + lane;
            const float ex0 = sE[0 * NEDGE + e];
            const float ey0 = sE[1 * NEDGE + e];
            const float edx = sE[2 * NEDGE + e];
            const float idy = sE[3 * NEDGE + e];
            const float cf  = sE[4 * NEDGE + e];
            const float t   = (yy - ey0) * idy;
            const float xc  = ex0 + t * edx;
            // valid_t = sig(20t)*sig(20-20t) with a single rcp
            const float ea  = __builtin_amdgcn_exp2f(-LOG2E * (20.0f * t));
            const float eb  = __builtin_amdgcn_exp2f(-LOG2E * (20.0f - 20.0f * t));
            const float wgt = cf * __builtin_amdgcn_rcpf((1.0f + ea) * (1.0f + eb));

            base += (xc >= xhi) ? wgt : 0.0f;       // saturated-high edges
            const bool active = (wgt != 0.0f) && (xc > xlo) && (xc < xhi);
            const unsigned m  = (unsigned)__ballot(active);
            if (active) {
                const int idx = cnt + __popc(m & ((1u << lane) - 1u));
                sAct[r][idx] = make_float2(xc, wgt);
            }
            cnt += __popc(m);
        }
        // deterministic wave-wide sum (fixed xor tree)
        #pragma unroll
        for (int off = 16; off > 0; off >>= 1)
            base += __shfl_xor(base, off, 32);
        if (lane == 0) { sBase[wv] = base; sCnt[wv] = cnt; }
    }
    __syncthreads();

    // ---- Phase 2: per-pixel accumulation over the compacted band ----
    const int   x  = x0blk + lane;
    const float fx = (float)x;
    float acc = sBase[wv];
    const int n = sCnt[wv];
    for (int i = 0; i < n; ++i) {
        const float2 p = sAct[wv][i];               // broadcast ds_load_b64
        acc += p.y * fast_sigmoid(p.x - fx);        // SIGMA_AA == 1.0
    }

    const float alpha = fast_sigmoid(4.0f * acc);
    out[(y0blk + wv) * W + x] = make_float4(color[0], color[1], color[2], alpha);
}

// ---------------------------------------------------------------------------
extern "C" void kernel_launch(void* const* d_in, const int* in_sizes, int n_in,
                              void* d_out, int out_size, void* d_ws, size_t ws_size,
                              hipStream_t stream) {
    const float* cp    = (const float*)d_in[0];   // (49,2) float32
    const float* color = (const float*)d_in[1];   // (3,)   float32
    float*  wsf  = (float*)d_ws;
    float*  pts  = wsf;                 // 1024 floats
    float*  edge = wsf + NPTS * 2;      // 2560 floats
    float4* out  = (float4*)d_out;      // (H,W,4) float32

    bezier_sample_wmma<<<1, 32, 0, stream>>>(cp, pts);
    build_edges<<<2, 256, 0, stream>>>(pts, edge);
    dim3 grid(W / TILE_X, H / TILE_Y);  // (32, 128)
    render_winding<<<grid, 256, 0, stream>>>(edge, color, out);
}